// binary_head_52544629899785
// MI455X (gfx1250) — compile-verified
//
#include <hip/hip_runtime.h>
#include <math.h>

// Problem constants (from reference): L=6, N=32, Q=100, C=256, H=8, Dh=32
#define N_  32
#define Q_  100
#define C_  256
#define H_  8
#define DH  32

typedef float v2f __attribute__((ext_vector_type(2)));
typedef float v8f __attribute__((ext_vector_type(8)));

// ---------------------------------------------------------------------------
// D[M x 256] = A[M x 256] @ W[256 x 256]^T + bias   (full f32 via WMMA)
// One wave32 per 16x64 output strip: 4 accumulators share one A fragment per
// K-step -> 4 v_wmma_f32_16x16x4_f32 per 5 b64 loads. K=256 -> 256 WMMAs/wave.
// A layout (16x4 f32): lanes 0-15 hold M=lane, K=k0..k0+1; lanes 16-31 K=k0+2..3.
// B layout (4x16 f32): lanes 0-15 hold N=lane, rows K=k0..k0+1; lanes 16-31 K+2.
// C/D layout (16x16 f32): lane column = lane&15, VGPR j = row (lane>=16 ? 8+j : j).
// ---------------------------------------------------------------------------
__global__ __launch_bounds__(32) void gemm_bias_wmma(
    const float* __restrict__ A, const float* __restrict__ W,
    const float* __restrict__ bias, float* __restrict__ D)
{
  const int lane = threadIdx.x;
  const int mt = blockIdx.x, nt = blockIdx.y;         // nt covers 64 columns
  const int half = lane >> 4;      // 0 -> K 0..1 / out rows 0..7 ; 1 -> K 2..3 / rows 8..15
  const int l15  = lane & 15;
  const float* arow = A + (size_t)(mt * 16 + l15) * C_ + half * 2;
  const float* w0 = W + (size_t)(nt * 64 +  0 + l15) * C_ + half * 2;  // W^T col = W row
  const float* w1 = W + (size_t)(nt * 64 + 16 + l15) * C_ + half * 2;
  const float* w2 = W + (size_t)(nt * 64 + 32 + l15) * C_ + half * 2;
  const float* w3 = W + (size_t)(nt * 64 + 48 + l15) * C_ + half * 2;
  v8f acc0 = {}, acc1 = {}, acc2 = {}, acc3 = {};
#pragma unroll 4
  for (int k0 = 0; k0 < C_; k0 += 4) {
    // prefetch one cacheline ahead on the A stream (global_prefetch_b8, WGP scope)
    __builtin_prefetch(arow + k0 + 32, 0, 3);
    v2f a  = *(const v2f*)(arow + k0);
    v2f b0 = *(const v2f*)(w0 + k0);
    v2f b1 = *(const v2f*)(w1 + k0);
    v2f b2 = *(const v2f*)(w2 + k0);
    v2f b3 = *(const v2f*)(w3 + k0);
    acc0 = __builtin_amdgcn_wmma_f32_16x16x4_f32(false, a, false, b0, (short)0, acc0, false, false);
    acc1 = __builtin_amdgcn_wmma_f32_16x16x4_f32(false, a, false, b1, (short)0, acc1, false, false);
    acc2 = __builtin_amdgcn_wmma_f32_16x16x4_f32(false, a, false, b2, (short)0, acc2, false, false);
    acc3 = __builtin_amdgcn_wmma_f32_16x16x4_f32(false, a, false, b3, (short)0, acc3, false, false);
  }
  float* drow = D + (size_t)(mt * 16 + half * 8) * C_ + nt * 64 + l15;
  const float bv0 = bias[nt * 64 +  0 + l15];
  const float bv1 = bias[nt * 64 + 16 + l15];
  const float bv2 = bias[nt * 64 + 32 + l15];
  const float bv3 = bias[nt * 64 + 48 + l15];
#pragma unroll
  for (int j = 0; j < 8; ++j) {
    drow[(size_t)j * C_ +  0] = acc0[j] + bv0;
    drow[(size_t)j * C_ + 16] = acc1[j] + bv1;
    drow[(size_t)j * C_ + 32] = acc2[j] + bv2;
    drow[(size_t)j * C_ + 48] = acc3[j] + bv3;
  }
}

// ---------------------------------------------------------------------------
// sum_all[n,c] = sum_q hs[n,q,c];  centers[n,0,c] = mean (== mean_pair)
// ---------------------------------------------------------------------------
__global__ __launch_bounds__(256) void centers_kernel(
    const float* __restrict__ hs, float* __restrict__ sum_all, float* __restrict__ centers)
{
  const int n = blockIdx.x, c = threadIdx.x;
  const float* base = hs + (size_t)n * Q_ * C_ + c;
  float s = 0.f;
  for (int q = 0; q < Q_; ++q) s += base[(size_t)q * C_];
  sum_all[n * C_ + c] = s;
  centers[(size_t)(n * 2) * C_ + c] = s * (1.0f / Q_);
}

// ---------------------------------------------------------------------------
// Per n: ||hs[n,s]||^2 (s<100), ||c1||^2 (slot 100); argmax_s ||c1-hs_s||^2 -> c2.
// ---------------------------------------------------------------------------
__global__ __launch_bounds__(128) void far1_kernel(
    const float* __restrict__ hs, float* __restrict__ centers,
    float* __restrict__ hs_norm)
{
  __shared__ float c1s[C_];
  __shared__ float dbuf[Q_];
  __shared__ float red[128];
  __shared__ int   sfar;
  const int n = blockIdx.x, tid = threadIdx.x;
  for (int c = tid; c < C_; c += 128) c1s[c] = centers[(size_t)(n * 2) * C_ + c];
  __syncthreads();
  red[tid] = c1s[tid] * c1s[tid] + c1s[tid + 128] * c1s[tid + 128];
  __syncthreads();
  for (int o = 64; o > 0; o >>= 1) { if (tid < o) red[tid] += red[tid + o]; __syncthreads(); }
  const float c1n = red[0];
  for (int s = tid; s < Q_; s += 128) {
    const float* row = hs + (size_t)(n * Q_ + s) * C_;
    float dot = 0.f, nrm = 0.f;
    for (int c = 0; c < C_; ++c) { float v = row[c]; dot += c1s[c] * v; nrm += v * v; }
    hs_norm[n * 101 + s] = nrm;
    float d = c1n + nrm - 2.f * dot;
    dbuf[s] = d > 0.f ? d : 0.f;
  }
  if (tid == 0) hs_norm[n * 101 + 100] = c1n;
  __syncthreads();
  if (tid == 0) {                     // first-occurrence argmax (matches jnp)
    float best = dbuf[0]; int bi = 0;
    for (int s = 1; s < Q_; ++s) if (dbuf[s] > best) { best = dbuf[s]; bi = s; }
    sfar = bi;
  }
  __syncthreads();
  const float* src = hs + (size_t)(n * Q_ + sfar) * C_;
  for (int c = tid; c < C_; c += 128) centers[(size_t)(n * 2 + 1) * C_ + c] = src[c];
}

// ---------------------------------------------------------------------------
// Branch center build: c1 (rm or md), masked-argmax farthest -> interleaved cen rows.
// S = 100 (rm) or 101 (md, row 100 = mean = centers[n,0]).
// ---------------------------------------------------------------------------
__global__ __launch_bounds__(128) void cen_build_kernel(
    const float* __restrict__ hs, const float* __restrict__ sum_all,
    const float* __restrict__ centers, const float* __restrict__ hs_norm,
    float* __restrict__ cen, int md)
{
  __shared__ float c1s[C_];
  __shared__ float dbuf[101];
  __shared__ float red[128];
  __shared__ int   sfar;
  const int b = blockIdx.x, n = b / Q_, q = b % Q_, tid = threadIdx.x;
  const int excl = (q + Q_ - 1) % Q_;
  const float* ex = hs + (size_t)(n * Q_ + excl) * C_;
  for (int c = tid; c < C_; c += 128) {
    const float sa = sum_all[n * C_ + c];
    float c1v;
    if (md) c1v = (sa - ex[c] + centers[(size_t)(n * 2) * C_ + c]) * (1.f / 100.f);
    else    c1v = (sa - ex[c]) * (1.f / 99.f);
    c1s[c] = c1v;
    cen[(size_t)(n * 2 * Q_ + 2 * q) * C_ + c] = c1v;
  }
  __syncthreads();
  red[tid] = c1s[tid] * c1s[tid] + c1s[tid + 128] * c1s[tid + 128];
  __syncthreads();
  for (int o = 64; o > 0; o >>= 1) { if (tid < o) red[tid] += red[tid + o]; __syncthreads(); }
  const float c1n = red[0];
  const int S = Q_ + md;
  for (int s = tid; s < S; s += 128) {
    const float* row = (s < Q_) ? hs + (size_t)(n * Q_ + s) * C_
                                : centers + (size_t)(n * 2) * C_;
    float dot = 0.f;
    for (int c = 0; c < C_; ++c) dot += c1s[c] * row[c];
    float d = c1n + hs_norm[n * 101 + s] - 2.f * dot;
    d = d > 0.f ? d : 0.f;
    if (s == excl) d = -3.4e38f;      // em mask before argmax
    dbuf[s] = d;
  }
  __syncthreads();
  if (tid == 0) {
    float best = dbuf[0]; int bi = 0;
    for (int s = 1; s < S; ++s) if (dbuf[s] > best) { best = dbuf[s]; bi = s; }
    sfar = bi;
  }
  __syncthreads();
  const float* src = (sfar < Q_) ? hs + (size_t)(n * Q_ + sfar) * C_
                                 : centers + (size_t)(n * 2) * C_;
  for (int c = tid; c < C_; c += 128) cen[(size_t)(n * 2 * Q_ + 2 * q + 1) * C_ + c] = src[c];
}

// ---------------------------------------------------------------------------
// Sigmoid-MHA core for one query row: scores -> sigmoid -> (mask) -> normalize
// -> attn@V.  card = mean_h rowsum (pre-eps).  asg (mode 0 only) = mean_h attn.
// K/V row s<100 from khs/vhs (hs projections); s==100 from khc/vhc (mean row proj).
// ---------------------------------------------------------------------------
__global__ __launch_bounds__(128) void attention_kernel(
    const float* __restrict__ qh,
    const float* __restrict__ khs, const float* __restrict__ vhs,
    const float* __restrict__ khc, const float* __restrict__ vhc,
    float* __restrict__ attn_out, float* __restrict__ card, float* __restrict__ asg,
    int T, int S, int masked)
{
  __shared__ float qrow[C_];
  __shared__ float w[H_ * 101];
  __shared__ float sums[H_];
  const int b = blockIdx.x, n = b / T, t = b % T, tid = threadIdx.x;
  const float* qr = qh + (size_t)(n * T + t) * C_;
  for (int c = tid; c < C_; c += 128) qrow[c] = qr[c];
  __syncthreads();
  const float scale = 0.17677669529663688f;   // 1/sqrt(Dh=32)
  for (int p = tid; p < H_ * S; p += 128) {
    const int h = p / S, s = p - h * S;
    const float* kr = (s < Q_) ? khs + (size_t)(n * Q_ + s) * C_
                               : khc + (size_t)(n * 2) * C_;
    const float* qq = qrow + h * DH;
    const float* kk = kr + h * DH;
    float dot = 0.f;
    for (int i = 0; i < DH; ++i) dot += qq[i] * kk[i];
    float a = 1.f / (1.f + expf(-dot * scale));
    if (masked && s == (t >> 1)) a = 0.f;     // mask column q for rows 2q,2q+1
    w[h * 101 + s] = a;
  }
  __syncthreads();
  if (tid < H_) {
    float s0 = 0.f;
    for (int s = 0; s < S; ++s) s0 += w[tid * 101 + s];
    sums[tid] = s0;
  }
  __syncthreads();
  if (tid == 0) {
    float cv = 0.f;
    for (int h = 0; h < H_; ++h) cv += sums[h];
    card[n * T + t] = cv * (1.f / H_);
  }
  if (asg) {
    for (int s = tid; s < S; s += 128) {
      float m = 0.f;
      for (int h = 0; h < H_; ++h) m += w[h * 101 + s];
      asg[(size_t)(n * T + t) * Q_ + s] = m * (1.f / H_);
    }
  }
  __syncthreads();
  for (int c = tid; c < C_; c += 128) {
    const int h = c >> 5;
    const float invn = 1.f / (sums[h] + 1e-4f);
    float acc = 0.f;
    for (int s = 0; s < S; ++s) {
      const float* vr = (s < Q_) ? vhs + (size_t)(n * Q_ + s) * C_
                                 : vhc + (size_t)(n * 2) * C_;
      acc += w[h * 101 + s] * vr[c];
    }
    attn_out[(size_t)(n * T + t) * C_ + c] = acc * invn;
  }
}

// ---------------------------------------------------------------------------
// smha#1 selection: s_idx/d_idx (first-occurrence argmin/argmax over 2) + cd/cs.
// ---------------------------------------------------------------------------
__global__ __launch_bounds__(256) void select1_kernel(
    const float* __restrict__ card1, const float* __restrict__ mf1,
    float* __restrict__ cd, float* __restrict__ cs,
    float* __restrict__ out_sidx, float* __restrict__ out_didx)
{
  const int n = blockIdx.x, c = threadIdx.x;
  const float c0 = card1[n * 2], c1v = card1[n * 2 + 1];
  const int si = (c0 <= c1v) ? 0 : 1;
  const int di = (c0 >= c1v) ? 0 : 1;
  if (c == 0) { out_sidx[n] = (float)si; out_didx[n] = (float)di; }
  cd[n * C_ + c] = mf1[(size_t)(n * 2 + di) * C_ + c];
  cs[n * C_ + c] = mf1[(size_t)(n * 2 + si) * C_ + c];
}

// ---------------------------------------------------------------------------
// shift[n,q] = ||cdx - cd|| + ||csx - cs||   (cdx/csx picked by cardinality)
// ---------------------------------------------------------------------------
__global__ __launch_bounds__(128) void shift_kernel(
    const float* __restrict__ cardb, const float* __restrict__ mfb,
    const float* __restrict__ cd, const float* __restrict__ cs,
    float* __restrict__ out)
{
  __shared__ float rd[128], rs[128];
  const int b = blockIdx.x, n = b / Q_, q = b % Q_, tid = threadIdx.x;
  const float c0 = cardb[n * 2 * Q_ + 2 * q], c1v = cardb[n * 2 * Q_ + 2 * q + 1];
  const int di = (c0 >= c1v) ? 0 : 1;
  const int si = (c0 <= c1v) ? 0 : 1;
  const float* cdx = mfb + (size_t)(n * 2 * Q_ + 2 * q + di) * C_;
  const float* csx = mfb + (size_t)(n * 2 * Q_ + 2 * q + si) * C_;
  const float* cdr = cd + n * C_;
  const float* csr = cs + n * C_;
  float sd = 0.f, ss = 0.f;
  for (int c = tid; c < C_; c += 128) {
    const float dd = cdx[c] - cdr[c]; sd += dd * dd;
    const float de = csx[c] - csr[c]; ss += de * de;
  }
  rd[tid] = sd; rs[tid] = ss; __syncthreads();
  for (int o = 64; o > 0; o >>= 1) {
    if (tid < o) { rd[tid] += rd[tid + o]; rs[tid] += rs[tid + o]; }
    __syncthreads();
  }
  if (tid == 0) out[n * Q_ + q] = sqrtf(rd[0] + 1e-12f) + sqrtf(rs[0] + 1e-12f);
}

// ---------------------------------------------------------------------------
extern "C" void kernel_launch(void* const* d_in, const int* in_sizes, int n_in,
                              void* d_out, int out_size, void* d_ws, size_t ws_size,
                              hipStream_t stream)
{
  (void)in_sizes; (void)n_in; (void)out_size; (void)ws_size;
  const float* hs_pair = (const float*)d_in[0];
  const float* ipw     = (const float*)d_in[1];
  const float* ipb     = (const float*)d_in[2];
  const float* ow      = (const float*)d_in[3];
  const float* ob      = (const float*)d_in[4];
  const float* hs = hs_pair + (size_t)5 * N_ * Q_ * C_;   // only last layer is live
  const float* wq = ipw;               const float* bq = ipb;
  const float* wk = ipw + C_ * C_;     const float* bk = ipb + C_;
  const float* wv = ipw + 2 * C_ * C_; const float* bv = ipb + 2 * C_;

  float* ws = (float*)d_ws;
  size_t o = 0;
  float* sum_all = ws + o; o += N_ * C_;
  float* centers = ws + o; o += N_ * 2 * C_;
  float* hs_norm = ws + o; o += N_ * 101;
  float* khs  = ws + o; o += (size_t)N_ * Q_ * C_;   // K proj of hs (shared by all 3 SMHA)
  float* vhs  = ws + o; o += (size_t)N_ * Q_ * C_;
  float* khc  = ws + o; o += N_ * 2 * C_;            // K proj of centers (row n*2 = mean)
  float* vhc  = ws + o; o += N_ * 2 * C_;
  float* qh1  = ws + o; o += N_ * 2 * C_;
  float* att1 = ws + o; o += N_ * 2 * C_;
  float* card1 = ws + o; o += N_ * 2;
  float* mf1  = ws + o; o += N_ * 2 * C_;
  float* cd   = ws + o; o += N_ * C_;
  float* cs   = ws + o; o += N_ * C_;
  float* cen   = ws + o; o += (size_t)N_ * 2 * Q_ * C_;  // reused by rm then md
  float* qhc   = ws + o; o += (size_t)N_ * 2 * Q_ * C_;
  float* attb  = ws + o; o += (size_t)N_ * 2 * Q_ * C_;
  float* cardb = ws + o; o += N_ * 2 * Q_;
  float* mfb   = ws + o; o += (size_t)N_ * 2 * Q_ * C_;

  float* out      = (float*)d_out;
  float* out_asg  = out;          // assignment (32,2,100) = 6400
  float* out_sidx = out + 6400;   // 32
  float* out_didx = out + 6432;   // 32
  float* out_rm   = out + 6464;   // 3200
  float* out_md   = out + 9664;   // 3200

  // ---- centers + shared K/V projections (computed once) ----
  centers_kernel<<<N_, 256, 0, stream>>>(hs, sum_all, centers);
  far1_kernel<<<N_, 128, 0, stream>>>(hs, centers, hs_norm);
  gemm_bias_wmma<<<dim3(N_ * Q_ / 16, 4), 32, 0, stream>>>(hs, wk, bk, khs);
  gemm_bias_wmma<<<dim3(N_ * Q_ / 16, 4), 32, 0, stream>>>(hs, wv, bv, vhs);
  gemm_bias_wmma<<<dim3(N_ * 2 / 16, 4), 32, 0, stream>>>(centers, wk, bk, khc);
  gemm_bias_wmma<<<dim3(N_ * 2 / 16, 4), 32, 0, stream>>>(centers, wv, bv, vhc);

  // ---- SMHA #1 on centers ----
  gemm_bias_wmma<<<dim3(N_ * 2 / 16, 4), 32, 0, stream>>>(centers, wq, bq, qh1);
  attention_kernel<<<N_ * 2, 128, 0, stream>>>(qh1, khs, vhs, khc, vhc,
                                               att1, card1, out_asg, 2, Q_, 0);
  gemm_bias_wmma<<<dim3(N_ * 2 / 16, 4), 32, 0, stream>>>(att1, ow, ob, mf1);
  select1_kernel<<<N_, 256, 0, stream>>>(card1, mf1, cd, cs, out_sidx, out_didx);

  // ---- branch rm (S=100) ----
  cen_build_kernel<<<N_ * Q_, 128, 0, stream>>>(hs, sum_all, centers, hs_norm, cen, 0);
  gemm_bias_wmma<<<dim3(N_ * 2 * Q_ / 16, 4), 32, 0, stream>>>(cen, wq, bq, qhc);
  attention_kernel<<<N_ * 2 * Q_, 128, 0, stream>>>(qhc, khs, vhs, khc, vhc,
                                                    attb, cardb, nullptr, 2 * Q_, Q_, 1);
  gemm_bias_wmma<<<dim3(N_ * 2 * Q_ / 16, 4), 32, 0, stream>>>(attb, ow, ob, mfb);
  shift_kernel<<<N_ * Q_, 128, 0, stream>>>(cardb, mfb, cd, cs, out_rm);

  // ---- branch md (S=101, row 100 = mean) ----
  cen_build_kernel<<<N_ * Q_, 128, 0, stream>>>(hs, sum_all, centers, hs_norm, cen, 1);
  gemm_bias_wmma<<<dim3(N_ * 2 * Q_ / 16, 4), 32, 0, stream>>>(cen, wq, bq, qhc);
  attention_kernel<<<N_ * 2 * Q_, 128, 0, stream>>>(qhc, khs, vhs, khc, vhc,
                                                    attb, cardb, nullptr, 2 * Q_, Q_ + 1, 1);
  gemm_bias_wmma<<<dim3(N_ * 2 * Q_ / 16, 4), 32, 0, stream>>>(attb, ow, ob, mfb);
  shift_kernel<<<N_ * Q_, 128, 0, stream>>>(cardb, mfb, cd, cs, out_md);
}